// DYMAG_33500744909064
// MI455X (gfx1250) — compile-verified
//
#include <hip/hip_runtime.h>
#include <hip/hip_bf16.h>

typedef __attribute__((ext_vector_type(2))) float    v2f;
typedef __attribute__((ext_vector_type(8))) float    v8f;
typedef __attribute__((ext_vector_type(4))) unsigned u32x4;
typedef __attribute__((ext_vector_type(8))) unsigned u32x8;

#define FEATS 26
#define T_SNAPS 6
#define K_HOPS 4
#define STEPS_PER_SNAP 10
#define DT 0.1f

// WMMA-GEMV tiling: TDM moves 16 x KT fp32 tiles of W into LDS.
#define KT 192                 // K-extent per tile (multiple of 64, divides 7488)
#define LROW 204               // padded LDS row stride in dwords: 192 + (192/64)*4
#define BUFDW (16 * LROW)      // dwords per LDS buffer (3264)
#define BUFBYTES (BUFDW * 4)   // 13056 B per buffer

static inline int cdiv(int a, int b) { return (a + b - 1) / b; }

// ---------------- degree / weights ----------------
__global__ __launch_bounds__(256) void deg_kernel(const int* __restrict__ dst,
                                                  float* __restrict__ deg, int E) {
    int e = blockIdx.x * blockDim.x + threadIdx.x;
    if (e < E) atomicAdd(&deg[dst[e]], 1.0f);
}

__global__ __launch_bounds__(256) void invsqrt_kernel(const float* __restrict__ deg,
                                                      float* __restrict__ inv, int N) {
    int i = blockIdx.x * blockDim.x + threadIdx.x;
    if (i < N) inv[i] = rsqrtf(fmaxf(deg[i], 1.0f));
}

__global__ __launch_bounds__(256) void edgew_kernel(const int* __restrict__ src,
                                                    const int* __restrict__ dst,
                                                    const float* __restrict__ inv,
                                                    float* __restrict__ w, int E) {
    int e = blockIdx.x * blockDim.x + threadIdx.x;
    if (e < E) w[e] = inv[src[e]] * inv[dst[e]];
}

// ---------------- edge scatter (SpMV): wave-per-edge, lane = feature --------
// 26 active lanes gather one contiguous 104B row (1-2 cachelines) and issue
// 26 contiguous f32 atomics; node state (3 MB) is L2-resident.
__global__ __launch_bounds__(256) void scatter_kernel(const float* __restrict__ hin,
                                                      float* __restrict__ hout,
                                                      const int* __restrict__ src,
                                                      const int* __restrict__ dst,
                                                      const float* __restrict__ w, int E) {
    int gw = (int)((blockIdx.x * blockDim.x + threadIdx.x) >> 5);  // global wave id = edge
    int lane = threadIdx.x & 31;
    if (gw >= E) return;
    int s = src[gw], d = dst[gw];                 // broadcast loads (uniform in wave)
    float we = w ? w[gw] : 1.0f;
    if (lane < FEATS) {
        float v = hin[(size_t)s * FEATS + lane];
        atomicAdd(&hout[(size_t)d * FEATS + lane], we * v);
    }
}

// ---------------- fused RK4 stage ----------------
__global__ __launch_bounds__(256) void rk4_stage(const float* __restrict__ tmp,
                                                 const float* __restrict__ yin,
                                                 float* __restrict__ h,
                                                 float* __restrict__ acc,
                                                 float* __restrict__ y,
                                                 int mode, int n) {
    int i = blockIdx.x * blockDim.x + threadIdx.x;
    if (i >= n) return;
    float k = tmp[i] - yin[i];
    if (mode == 0)      { acc[i] = k;           y[i] = h[i] + (0.5f * DT) * k; }
    else if (mode == 1) { acc[i] += 2.0f * k;   y[i] = h[i] + (0.5f * DT) * k; }
    else if (mode == 2) { acc[i] += 2.0f * k;   y[i] = h[i] + DT * k;          }
    else                { h[i] += (DT / 6.0f) * (acc[i] + k);                  }
}

// ---------------- graph moments: g[t,s,k,m,f] = sum_n h[n,f]^((m+1)(s+1)) ---
__global__ __launch_bounds__(256) void moments_kernel(const float* __restrict__ hk,
                                                      float* __restrict__ g,
                                                      int t, int k, int N) {
    int f = blockIdx.x;
    int tid = threadIdx.x;
    float a[12];
#pragma unroll
    for (int j = 0; j < 12; ++j) a[j] = 0.0f;
    for (int n = tid; n < N; n += 256) {
        float p1 = hk[(size_t)n * FEATS + f];
        float p2 = p1 * p1, p3 = p2 * p1, p4 = p2 * p2;
        float p6 = p3 * p3, p8 = p4 * p4, p9 = p6 * p3, p12 = p6 * p6;
        a[0] += p1; a[1] += p2;  a[2] += p3;  a[3] += p4;
        a[4] += p2; a[5] += p4;  a[6] += p6;  a[7] += p8;
        a[8] += p3; a[9] += p6;  a[10] += p9; a[11] += p12;
    }
    __shared__ float sdata[256];
    for (int j = 0; j < 12; ++j) {
        sdata[tid] = a[j];
        __syncthreads();
        for (int s2 = 128; s2 > 0; s2 >>= 1) {
            if (tid < s2) sdata[tid] += sdata[tid + s2];
            __syncthreads();
        }
        if (tid == 0) {
            int s_ = j / 4, m_ = j % 4;
            g[((((t * 3 + s_) * K_HOPS + k) * 4 + m_) * FEATS) + f] = sdata[0];
        }
        __syncthreads();
    }
}

// ---------------- TDM descriptor build + issue (wave0 only) -----------------
// D# group0/group1 per CDNA5 ISA §8: 2D tile, fp32 (data_size=2 -> 4B),
// tile = KT x rows, tensor row stride = K elements. LDS padding: 4 dwords
// every 64 dwords (pad_interval code 5, pad_amount field 3) -> LDS row stride
// 204 dwords, giving conflict-free ds_load_b64 of A fragments.
__device__ __forceinline__ void tdm_issue_tile(unsigned lds_addr, unsigned long long gaddr,
                                               int K, int rows) {
    u32x4 g0;
    g0[0] = 1u;                                              // count=1, user mode
    g0[1] = lds_addr;                                        // LDS byte address
    g0[2] = (unsigned)(gaddr & 0xFFFFFFFFu);                 // global_addr[31:0]
    g0[3] = (unsigned)((gaddr >> 32) & 0x1FFFFFFu) | (2u << 30);  // [56:32] | type=2
    u32x8 g1;
    g1[0] = (2u << 16)      // data_size = 4B
          | (1u << 20)      // pad_enable
          | (5u << 22)      // pad_interval: 64 dwords
          | (3u << 25);     // pad_amount: 4 dwords
    g1[1] = ((unsigned)K & 0xFFFFu) << 16;                   // tensor_dim0[15:0]
    g1[2] = (((unsigned)K >> 16) & 0xFFFFu)                  // tensor_dim0[31:16]
          | (((unsigned)rows & 0xFFFFu) << 16);              // tensor_dim1[15:0]
    g1[3] = ((unsigned)KT & 0xFFFFu) << 16;                  // tile_dim0
    g1[4] = (unsigned)rows & 0xFFFFu;                        // tile_dim1
    g1[5] = (unsigned)K;                                     // tensor_dim0_stride[31:0]
    g1[6] = 0u;
    g1[7] = 0u;
    asm volatile("tensor_load_to_lds %0, %1" :: "s"(g0), "s"(g1) : "memory");
}

// ---------------- WMMA GEMV with TDM double-buffered W tiles ----------------
// One block (8 waves) per 16-row tile of W. TDM streams 16xKT fp32 tiles of W
// into LDS; each wave covers KT/8 of the tile's K with V_WMMA_F32_16X16X4_F32.
// B columns are replicated with the v-chunk so every D column holds the GEMV
// partial; lanes 0/16 carry the 16 row results; LDS-reduce across waves.
__global__ __launch_bounds__(256) void wmma_gemv(const float* __restrict__ W,
                                                 const float* __restrict__ bias,
                                                 const float* __restrict__ vin,
                                                 float* __restrict__ vout,
                                                 int M, int K, int relu) {
    __shared__ __align__(16) float lds[2 * BUFDW];   // 2 x 13056 B
    __shared__ float red[8][16];
    int tid = threadIdx.x;
    int wave = tid >> 5, lane = tid & 31;
    int hi = lane >> 4, mr = lane & 15;
    int row0 = blockIdx.x * 16;
    int rows = M - row0; if (rows > 16) rows = 16;
    int nt = K / KT;                                  // 39 for K=7488

    unsigned lds_base = (unsigned)(uintptr_t)&lds[0];
    unsigned long long gbase = (unsigned long long)(uintptr_t)W
                             + (unsigned long long)row0 * (unsigned)K * 4ull;

    if (wave == 0)
        tdm_issue_tile(lds_base, gbase, K, rows);

    v8f c = {};
    for (int i = 0; i < nt; ++i) {
        if (wave == 0) {
            if (i + 1 < nt) {
                tdm_issue_tile(lds_base + (unsigned)(((i + 1) & 1) * BUFBYTES),
                               gbase + (unsigned long long)(i + 1) * KT * 4ull,
                               K, rows);
                asm volatile("s_wait_tensorcnt 0x1" ::: "memory");  // tile i done
            } else {
                asm volatile("s_wait_tensorcnt 0x0" ::: "memory");
            }
        }
        __syncthreads();                              // tile i visible to all waves
        const float* buf = &lds[(i & 1) * BUFDW];
        int kb = i * KT;
        int kl0 = wave * (KT / 8);
#pragma unroll
        for (int u = 0; u < KT / 32; ++u) {           // KT/8 per wave, step 4
            int kl = kl0 + u * 4;
            // A fragment from LDS: lane=M row, K = vgpr + 2*hi ; padded row map
            const float* ap = &buf[(size_t)mr * LROW + kl + 2 * hi + 4 * (kl >> 6)];
            float2 av = *(const float2*)ap;           // ds_load_b64, conflict-free
            float2 bv = *(const float2*)(vin + kb + kl + 2 * hi);
            v2f a, b;
            a[0] = av.x; a[1] = av.y;
            b[0] = bv.x; b[1] = bv.y;                 // all N columns identical
            c = __builtin_amdgcn_wmma_f32_16x16x4_f32(false, a, false, b,
                                                      (short)0, c, false, false);
        }
        __syncthreads();                              // done reading buf[i&1]
    }

    if (mr == 0) {
#pragma unroll
        for (int r = 0; r < 8; ++r) red[wave][r + 8 * hi] = c[r];
    }
    __syncthreads();
    if (tid < 16) {
        int orow = row0 + tid;
        if (orow < M) {
            float s = 0.0f;
#pragma unroll
            for (int wv = 0; wv < 8; ++wv) s += red[wv][tid];
            s += bias[orow];
            if (relu) s = fmaxf(s, 0.0f);
            vout[orow] = s;
        }
    }
}

// ---------------- host orchestration ----------------
extern "C" void kernel_launch(void* const* d_in, const int* in_sizes, int n_in,
                              void* d_out, int out_size, void* d_ws, size_t ws_size,
                              hipStream_t stream) {
    (void)n_in; (void)out_size; (void)ws_size;
    const float* x   = (const float*)d_in[0];
    const int*   ei  = (const int*)d_in[1];
    const float* W1  = (const float*)d_in[2];
    const float* b1  = (const float*)d_in[3];
    const float* W2  = (const float*)d_in[4];
    const float* b2  = (const float*)d_in[5];
    const float* Wc  = (const float*)d_in[6];
    const float* bc  = (const float*)d_in[7];
    float* out = (float*)d_out;

    const int N = in_sizes[0] / FEATS;       // 30000
    const int E = in_sizes[1] / 2;           // 480000
    const int H = in_sizes[3];               // 7488
    const int NOUT = in_sizes[7];            // 26
    const size_t NF = (size_t)N * FEATS;

    const int* src = ei;
    const int* dst = ei + E;

    float* p = (float*)d_ws;
    float* deg  = p;  p += N;
    float* inv  = p;  p += N;
    float* wbuf = p;  p += E;
    float* h    = p;  p += NF;
    float* y    = p;  p += NF;
    float* tmp  = p;  p += NF;
    float* acc  = p;  p += NF;
    float* hk   = p;  p += NF;
    float* hk2  = p;  p += NF;
    float* g    = p;  p += 7680;
    float* v1   = p;  p += 7680;
    float* v2   = p;  p += 7680;

    const int gE   = cdiv(E, 256);           // thread-per-edge kernels
    const int gEw  = cdiv(E, 8);             // wave-per-edge scatter (8 waves/block)
    const int gN   = cdiv(N, 256);
    const int gNF  = cdiv((int)NF, 256);

    // sym-normalized edge weights
    hipMemsetAsync(deg, 0, (size_t)N * sizeof(float), stream);
    deg_kernel<<<gE, 256, 0, stream>>>(dst, deg, E);
    invsqrt_kernel<<<gN, 256, 0, stream>>>(deg, inv, N);
    edgew_kernel<<<gE, 256, 0, stream>>>(src, dst, inv, wbuf, E);

    // h = x
    hipMemcpyAsync(h, x, NF * sizeof(float), hipMemcpyDeviceToDevice, stream);

    for (int t = 0; t < T_SNAPS; ++t) {
        // ---- k-hop moments of current snapshot ----
        hipMemcpyAsync(hk, h, NF * sizeof(float), hipMemcpyDeviceToDevice, stream);
        float* a0 = hk; float* a1 = hk2;
        for (int k = 0; k < K_HOPS; ++k) {
            hipMemsetAsync(a1, 0, NF * sizeof(float), stream);
            scatter_kernel<<<gEw, 256, 0, stream>>>(a0, a1, src, dst, nullptr, E);
            float* sw = a0; a0 = a1; a1 = sw;
            moments_kernel<<<FEATS, 256, 0, stream>>>(a0, g, t, k, N);
        }
        // ---- advance 10 RK4 steps ----
        if (t < T_SNAPS - 1) {
            for (int s = 0; s < STEPS_PER_SNAP; ++s) {
                hipMemsetAsync(tmp, 0, NF * sizeof(float), stream);
                scatter_kernel<<<gEw, 256, 0, stream>>>(h, tmp, src, dst, wbuf, E);
                rk4_stage<<<gNF, 256, 0, stream>>>(tmp, h, h, acc, y, 0, (int)NF);
                for (int m = 1; m < 4; ++m) {
                    hipMemsetAsync(tmp, 0, NF * sizeof(float), stream);
                    scatter_kernel<<<gEw, 256, 0, stream>>>(y, tmp, src, dst, wbuf, E);
                    rk4_stage<<<gNF, 256, 0, stream>>>(tmp, y, h, acc, y, m, (int)NF);
                }
            }
        }
    }

    // ---- MLP head (WMMA GEMVs with TDM-staged weight tiles) ----
    wmma_gemv<<<cdiv(H, 16), 256, 0, stream>>>(W1, b1, g,  v1, H, H, 1);
    wmma_gemv<<<cdiv(H, 16), 256, 0, stream>>>(W2, b2, v1, v2, H, H, 1);
    wmma_gemv<<<cdiv(NOUT, 16), 256, 0, stream>>>(Wc, bc, v2, out, NOUT, H, 0);
}